// AbstractNN_40965398069641
// MI455X (gfx1250) — compile-verified
//
#include <hip/hip_runtime.h>

typedef __attribute__((ext_vector_type(2))) float v2f;
typedef __attribute__((ext_vector_type(8))) float v8f;

#define D_DIM 2048
#define BM 128
#define BNB 128
#define BK 16
#define LDA 20   // padded LDS stride (multiple of 4 -> 16B-aligned b128 stores)

#if defined(__has_builtin)
#if __has_builtin(__builtin_amdgcn_global_load_async_to_lds_b128)
#define USE_ASYNC_LDS 1
#endif
#endif

#ifdef USE_ASYNC_LDS
// Builtin signature (probe-confirmed): (int4 __device__* src, int4 LDS* dst, imm offset, imm cpol)
typedef int v4i_b __attribute__((vector_size(16)));
typedef v4i_b __attribute__((address_space(1)))* async_gptr;
typedef v4i_b __attribute__((address_space(3)))* async_lptr;

__device__ __forceinline__ void wait_async_le0() {
#if __has_builtin(__builtin_amdgcn_s_wait_asynccnt)
  __builtin_amdgcn_s_wait_asynccnt(0);
#else
  asm volatile("s_wait_asynccnt 0x0" ::: "memory");
#endif
}
__device__ __forceinline__ void wait_async_le4() {
#if __has_builtin(__builtin_amdgcn_s_wait_asynccnt)
  __builtin_amdgcn_s_wait_asynccnt(4);
#else
  asm volatile("s_wait_asynccnt 0x4" ::: "memory");
#endif
}
__device__ __forceinline__ void issue_tile_async(const float* ag, const float* bg,
                                                 float* al, float* bl) {
  __builtin_amdgcn_global_load_async_to_lds_b128((async_gptr)ag,       (async_lptr)al,       0, 0);
  __builtin_amdgcn_global_load_async_to_lds_b128((async_gptr)(ag + 4), (async_lptr)(al + 4), 0, 0);
  __builtin_amdgcn_global_load_async_to_lds_b128((async_gptr)bg,       (async_lptr)bl,       0, 0);
  __builtin_amdgcn_global_load_async_to_lds_b128((async_gptr)(bg + 4), (async_lptr)(bl + 4), 0, 0);
}
#endif

#define WMMA_F32(A_, B_, C_) \
  __builtin_amdgcn_wmma_f32_16x16x4_f32(false, (A_), false, (B_), (short)0, (C_), false, false)

// Compute 32 WMMAs (2 M-tiles x 4 N-tiles x 4 K-chunks) from one staged tile.
__device__ __forceinline__ void compute_tile(
    const float* __restrict__ AsB, const float* __restrict__ BsB,
    int wm, int wn, int fcol, int fk,
    v8f& c00, v8f& c01, v8f& c02, v8f& c03,
    v8f& c10, v8f& c11, v8f& c12, v8f& c13) {
#pragma unroll
  for (int kc = 0; kc < BK; kc += 4) {
    const int fo = kc + fk;
    v2f a0 = *(const v2f*)(AsB + (wm + fcol) * LDA + fo);
    v2f a1 = *(const v2f*)(AsB + (wm + 16 + fcol) * LDA + fo);
    v2f b0 = *(const v2f*)(BsB + (wn + fcol) * LDA + fo);
    v2f b1 = *(const v2f*)(BsB + (wn + 16 + fcol) * LDA + fo);
    v2f b2 = *(const v2f*)(BsB + (wn + 32 + fcol) * LDA + fo);
    v2f b3 = *(const v2f*)(BsB + (wn + 48 + fcol) * LDA + fo);
    c00 = WMMA_F32(a0, b0, c00);
    c01 = WMMA_F32(a0, b1, c01);
    c02 = WMMA_F32(a0, b2, c02);
    c03 = WMMA_F32(a0, b3, c03);
    c10 = WMMA_F32(a1, b0, c10);
    c11 = WMMA_F32(a1, b1, c11);
    c12 = WMMA_F32(a1, b2, c12);
    c13 = WMMA_F32(a1, b3, c13);
  }
}

// ---------------------------------------------------------------------------
// WMMA f32 GEMM: Y[m, n] = A[m, :] . W[n, :]  (+ bias per bias_mode)
// bias_mode: 0 = none, 1 = row 0 only, 2 = all rows
// Workgroup tile 128x128; 8 waves, each owning a 32x64 register tile.
// Double-buffered async global->LDS staging (ASYNCcnt pipelined).
// ---------------------------------------------------------------------------
__global__ __launch_bounds__(256) void gemm_wmma_f32(
    const float* __restrict__ A, const float* __restrict__ W,
    const float* __restrict__ bias, float* __restrict__ Y,
    int M_A /*readable A rows*/, int M_store /*rows to write*/, int bias_mode) {
  __shared__ float As[2][BM * LDA];
  __shared__ float Bs[2][BNB * LDA];

  const int t    = threadIdx.x;
  const int lane = t & 31;
  const int wave = t >> 5;
  const int wm   = (wave & 3) * 32;   // 4 waves down  -> 128 rows
  const int wn   = (wave >> 2) * 64;  // 2 waves across -> 128 cols

  const long long mBlock = (long long)blockIdx.x * BM;
  const long long nBlock = (long long)blockIdx.y * BNB;

  v8f c00{}, c01{}, c02{}, c03{}, c10{}, c11{}, c12{}, c13{};

  // Staging: each thread moves 8 floats of A and 8 floats of B per BK step.
  const int srow = t >> 1;
  const int scol = (t & 1) * 8;
  long long ar = mBlock + srow;
  if (ar >= M_A) ar = M_A - 1;        // clamp (rows beyond M_store never stored)
  const float* aptr = A + ar * D_DIM + scol;
  const float* bptr = W + (nBlock + srow) * D_DIM + scol;
  const int soff = srow * LDA + scol;

  const int fcol = lane & 15;         // row-in-tile for fragments
  const int fk   = (lane >> 4) << 1;  // lanes 16-31 hold K+2,K+3

#ifdef USE_ASYNC_LDS
  // Prologue: tile 0 -> buffer 0
  issue_tile_async(aptr, bptr, &As[0][soff], &Bs[0][soff]);

  for (int kb = 0; kb < D_DIM; kb += BK) {
    const int cur = (kb >> 4) & 1;
    __syncthreads();   // all waves done reading buffer cur^1 (previous compute)
    if (kb + BK < D_DIM) {
      issue_tile_async(aptr + kb + BK, bptr + kb + BK,
                       &As[cur ^ 1][soff], &Bs[cur ^ 1][soff]);
      wait_async_le4();   // tile kb's 4 copies complete; kb+BK still in flight
    } else {
      wait_async_le0();
    }
    __syncthreads();   // tile kb visible to all waves
    compute_tile(&As[cur][0], &Bs[cur][0], wm, wn, fcol, fk,
                 c00, c01, c02, c03, c10, c11, c12, c13);
  }
#else
  for (int kb = 0; kb < D_DIM; kb += BK) {
    float4 av0 = *(const float4*)(aptr + kb);
    float4 av1 = *(const float4*)(aptr + kb + 4);
    float4 bv0 = *(const float4*)(bptr + kb);
    float4 bv1 = *(const float4*)(bptr + kb + 4);
    __syncthreads();
    *(float4*)(&As[0][soff])     = av0;
    *(float4*)(&As[0][soff + 4]) = av1;
    *(float4*)(&Bs[0][soff])     = bv0;
    *(float4*)(&Bs[0][soff + 4]) = bv1;
    __syncthreads();
    compute_tile(&As[0][0], &Bs[0][0], wm, wn, fcol, fk,
                 c00, c01, c02, c03, c10, c11, c12, c13);
  }
#endif

  // Store: C/D layout — lane<16: rows m0+v, lane>=16: rows m0+8+v; col = n0+(lane&15)
  const int rowSel = (lane >> 4) * 8;
  v8f accs[2][4] = {{c00, c01, c02, c03}, {c10, c11, c12, c13}};
#pragma unroll
  for (int mt = 0; mt < 2; ++mt) {
#pragma unroll
    for (int nt = 0; nt < 4; ++nt) {
      v8f acc = accs[mt][nt];
      long long gc = nBlock + wn + nt * 16 + fcol;
      float bv = (bias_mode == 2) ? bias[gc] : 0.0f;
#pragma unroll
      for (int v = 0; v < 8; ++v) {
        long long gr = mBlock + wm + mt * 16 + v + rowSel;
        if (gr < M_store) {
          float val = acc[v] + bv;
          if (bias_mode == 1 && gr == 0) val += bias[gc];
          Y[gr * D_DIM + gc] = val;
        }
      }
    }
  }
}

// ---------------------------------------------------------------------------
// Zero-fill (float4 granularity)
// ---------------------------------------------------------------------------
__global__ void zero_f4(float4* __restrict__ p, long long n4) {
  long long i = (long long)blockIdx.x * blockDim.x + threadIdx.x;
  if (i < n4) p[i] = make_float4(0.f, 0.f, 0.f, 0.f);
}

// ---------------------------------------------------------------------------
// One-hot scatters
// ---------------------------------------------------------------------------
__global__ void scatter_w_kernel(const float* __restrict__ xc,
                                 const float* __restrict__ w_val,
                                 const int* __restrict__ w_idx,
                                 float* __restrict__ out, int nw, long long row0) {
  int n = blockIdx.x * blockDim.x + threadIdx.x;
  if (n < nw) {
    int idx = w_idx[n];
    int col = idx >> 11;        // idx / 2048 -> output column
    int cin = idx & 2047;       // idx % 2048 -> input column
    out[(row0 + n) * D_DIM + col] = w_val[n] * xc[cin];
  }
}

__global__ void scatter_b_kernel(const float* __restrict__ b_val,
                                 const int* __restrict__ b_idx,
                                 float* __restrict__ out, int nb, long long row0) {
  int n = blockIdx.x * blockDim.x + threadIdx.x;
  if (n < nb) out[(row0 + n) * D_DIM + b_idx[n]] = b_val[n];
}

// ---------------------------------------------------------------------------
// Noise row: out[col] = sum_k xn[k] * |W[col,k]|  — one wave per column
// ---------------------------------------------------------------------------
__global__ void noise_kernel(const float* __restrict__ xn,
                             const float* __restrict__ W,
                             float* __restrict__ out) {
  int lane = threadIdx.x & 31;
  int wv   = threadIdx.x >> 5;
  int col  = blockIdx.x * (blockDim.x >> 5) + wv;
  float s = 0.f;
  for (int k = lane; k < D_DIM; k += 32)
    s += xn[k] * fabsf(W[(long long)col * D_DIM + k]);
#pragma unroll
  for (int off = 16; off > 0; off >>= 1) s += __shfl_xor(s, off, 32);
  if (lane == 0) out[col] = s;
}

// ---------------------------------------------------------------------------
// Column abs-sum over eps rows (rows 1..8192), 64 deterministic chunks
// ---------------------------------------------------------------------------
__global__ void abs_stage1(const float* __restrict__ Y, float* __restrict__ partials) {
  int c = blockIdx.x * blockDim.x + threadIdx.x;   // column
  int chunk = blockIdx.y;                           // 0..63, 128 rows each
  long long r0 = 1 + (long long)chunk * 128;
  float s = 0.f;
  for (int i = 0; i < 128; ++i)
    s += fabsf(Y[(r0 + i) * D_DIM + c]);
  partials[(long long)chunk * D_DIM + c] = s;
}

// ---------------------------------------------------------------------------
// Final: fold partials + |noise| + scanned one-hot contributions -> min/max
// ---------------------------------------------------------------------------
__global__ void minmax_kernel(const float* __restrict__ Y,
                              const float* __restrict__ partials, int nchunks,
                              const float* __restrict__ xc,
                              const float* __restrict__ w_val, const int* __restrict__ w_idx,
                              const float* __restrict__ b_val, const int* __restrict__ b_idx,
                              float* __restrict__ xmin, float* __restrict__ xmax,
                              int nw, int nb, long long noise_row) {
  int c = blockIdx.x * blockDim.x + threadIdx.x;
  float s = 0.f;
  for (int i = 0; i < nchunks; ++i) s += partials[(long long)i * D_DIM + c];
  s += fabsf(Y[noise_row * D_DIM + c]);
  for (int n = 0; n < nw; ++n) {
    int idx = w_idx[n];
    if ((idx >> 11) == c) s += fabsf(w_val[n] * xc[idx & 2047]);
  }
  for (int n = 0; n < nb; ++n)
    if (b_idx[n] == c) s += fabsf(b_val[n]);
  float v0 = Y[c];   // x_out[0] = x_value (with bias)
  xmin[c] = v0 - s;
  xmax[c] = v0 + s;
}

// ---------------------------------------------------------------------------
extern "C" void kernel_launch(void* const* d_in, const int* in_sizes, int n_in,
                              void* d_out, int out_size, void* d_ws, size_t ws_size,
                              hipStream_t stream) {
  const float* x     = (const float*)d_in[0];  // [8194, 2048]
  const float* xtrue = (const float*)d_in[1];  // [32, 2048]
  const float* W     = (const float*)d_in[2];  // [2048, 2048]
  const float* bvec  = (const float*)d_in[3];  // [2048]
  const float* w_val = (const float*)d_in[4];  // [NW]
  const float* b_val = (const float*)d_in[5];  // [NB]
  const int*   w_idx = (const int*)d_in[6];
  const int*   b_idx = (const int*)d_in[7];
  float* out = (float*)d_out;

  const int NW = in_sizes[4];                  // 4096
  const int NB = in_sizes[5];                  // 2048
  const int MT = in_sizes[1] / D_DIM;          // 32
  const int EPS_STORE = 1 + 8192;              // rows 0..8192 from GEMM
  const long long xout_rows = 1LL + 8192 + NW + NB + 1;  // 14338
  const long long row_w0 = EPS_STORE;          // 8193
  const long long row_b0 = row_w0 + NW;        // 12289
  const long long noise_row = row_b0 + NB;     // 14337

  float* xmin  = out + xout_rows * D_DIM;
  float* xmax  = xmin + D_DIM;
  float* ytrue = xmax + D_DIM;
  float* partials = (float*)d_ws;              // 64 * 2048 floats

  dim3 blk(256);

  // 1) Main GEMM: x rows 0..8192 -> x_out rows 0..8192 (bias on row 0 only)
  dim3 g1((EPS_STORE + BM - 1) / BM, D_DIM / BNB);
  gemm_wmma_f32<<<g1, blk, 0, stream>>>(x, W, bvec, out, 8194, EPS_STORE, 1);

  // 2) x_true GEMM (bias on all rows)
  dim3 g2((MT + BM - 1) / BM, D_DIM / BNB);
  gemm_wmma_f32<<<g2, blk, 0, stream>>>(xtrue, W, bvec, ytrue, MT, MT, 2);

  // 3) Zero add_w / add_b block
  long long n4 = (long long)(NW + NB) * D_DIM / 4;
  zero_f4<<<(unsigned)((n4 + 255) / 256), blk, 0, stream>>>(
      (float4*)(out + row_w0 * D_DIM), n4);

  // 4) One-hot scatters
  scatter_w_kernel<<<(NW + 255) / 256, blk, 0, stream>>>(x, w_val, w_idx, out, NW, row_w0);
  scatter_b_kernel<<<(NB + 255) / 256, blk, 0, stream>>>(b_val, b_idx, out, NB, row_b0);

  // 5) Noise row: x[-1] @ |W|.T
  noise_kernel<<<D_DIM / 8, blk, 0, stream>>>(x + 8193LL * D_DIM, W, out + noise_row * D_DIM);

  // 6) Column abs-sum partials over eps rows 1..8192
  dim3 g6(D_DIM / 256, 64);
  abs_stage1<<<g6, blk, 0, stream>>>(out, partials);

  // 7) Fold everything into x_min / x_max
  minmax_kernel<<<D_DIM / 256, blk, 0, stream>>>(out, partials, 64, x, w_val, w_idx,
                                                 b_val, b_idx, xmin, xmax, NW, NB, noise_row);
}